// RNN_44779329028135
// MI455X (gfx1250) — compile-verified
//
#include <hip/hip_runtime.h>
#include <math.h>

// ---------------------------------------------------------------------------
// RNN scan -> parallel form (h-update is purely additive => prefix sum):
//   inter2_t = f(a_t, z_t)             (batched WMMA GEMMs L1,L2)
//   u_t      = g(inter2_t)             (batched WMMA GEMMs L3,L4)
//   h_t      = prefix_sum(u)_t         (hierarchical scan)
//   losses   = head(h_t, inter2_t)     (batched WMMA GEMMs L5,Lz + dot Lt)
// All GEMMs: V_WMMA_F32_16X16X4_F32 (wave32, fp32-exact vs reference).
// Weights/biases are zero-padded once into workspace so the WMMA inner loops
// are pure unconditional ds_load_b64 / global_load_b64 + v_wmma (no exec
// juggling, no per-element guards).
// ---------------------------------------------------------------------------

typedef __attribute__((ext_vector_type(2))) float v2f;
typedef __attribute__((ext_vector_type(8))) float v8f;

#define T_TOT   16384
#define S_STEPS 16383
#define SLOPE   0.01f

#define KP1 104   // az (102) padded
#define NP  112   // 100-dim padded
#define KP5 176   // [h(50)->64 | inter(100)->112] padded
#define NP4 64    // 50-dim padded

// pad-region offsets in floats (all multiples of 16 -> 64B aligned)
#define O_W1P 0
#define O_W2P (O_W1P + NP * KP1)
#define O_W3P (O_W2P + NP * NP)
#define O_W4P (O_W3P + NP * NP)
#define O_W5P (O_W4P + NP4 * NP)
#define O_WZP (O_W5P + NP * KP5)
#define O_WTP (O_WZP + NP * NP)
#define O_B1P (O_WTP + NP)
#define O_B2P (O_B1P + NP)
#define O_B3P (O_B2P + NP)
#define O_B4P (O_B3P + NP)
#define O_B5P (O_B4P + NP4)
#define O_BZP (O_B5P + NP)
#define PAD_TOTAL (O_BZP + NP)

// ---- one-time zero-padded weight/bias staging -----------------------------

__global__ __launch_bounds__(256) void k_pad_all(
    const float* __restrict__ W1, const float* __restrict__ W2,
    const float* __restrict__ W3, const float* __restrict__ W4,
    const float* __restrict__ W5, const float* __restrict__ Wz,
    const float* __restrict__ Wt, const float* __restrict__ b1,
    const float* __restrict__ b2, const float* __restrict__ b3,
    const float* __restrict__ b4, const float* __restrict__ b5,
    const float* __restrict__ bz, float* __restrict__ pad)
{
  int i = blockIdx.x * 256 + threadIdx.x;
  if (i >= PAD_TOTAL) return;
  float v = 0.f;
  if (i < O_W2P) {                         // W1p [112][104] <- [100][102]
    int j = i - O_W1P, n = j / KP1, k = j % KP1;
    if (n < 100 && k < 102) v = W1[n * 102 + k];
  } else if (i < O_W3P) {                  // W2p [112][112] <- [100][100]
    int j = i - O_W2P, n = j / NP, k = j % NP;
    if (n < 100 && k < 100) v = W2[n * 100 + k];
  } else if (i < O_W4P) {                  // W3p
    int j = i - O_W3P, n = j / NP, k = j % NP;
    if (n < 100 && k < 100) v = W3[n * 100 + k];
  } else if (i < O_W5P) {                  // W4p [64][112] <- [50][100]
    int j = i - O_W4P, n = j / NP, k = j % NP;
    if (n < 50 && k < 100) v = W4[n * 100 + k];
  } else if (i < O_WZP) {                  // W5p [112][176], gap mapping baked
    int j = i - O_W5P, n = j / KP5, k = j % KP5;
    if (n < 100) {
      if (k < 50)                   v = W5[n * 150 + k];
      else if (k >= 64 && k < 164)  v = W5[n * 150 + (k - 14)];
    }
  } else if (i < O_WTP) {                  // Wzp [112][112]
    int j = i - O_WZP, n = j / NP, k = j % NP;
    if (n < 100 && k < 100) v = Wz[n * 100 + k];
  } else if (i < O_B1P) { int j = i - O_WTP; if (j < 100) v = Wt[j]; }
  else if (i < O_B2P)   { int j = i - O_B1P; if (j < 100) v = b1[j]; }
  else if (i < O_B3P)   { int j = i - O_B2P; if (j < 100) v = b2[j]; }
  else if (i < O_B4P)   { int j = i - O_B3P; if (j < 100) v = b3[j]; }
  else if (i < O_B5P)   { int j = i - O_B4P; if (j < 50)  v = b4[j]; }
  else if (i < O_BZP)   { int j = i - O_B5P; if (j < 100) v = b5[j]; }
  else                  { int j = i - O_BZP; if (j < 100) v = bz[j]; }
  pad[i] = v;
}

// ---- WMMA helpers (ISA 7.12.2 fp32 layouts, wave32) -----------------------
// A 16x4: vgpr0/1 = K {k0,k0+1} (lanes 0-15) / {k0+2,k0+3} (lanes 16-31)
// B 4x16: same K split, N = lane&15;  C/D: vgpr r = row r / r+8.
// kb is always even and rows are 8B-aligned -> single b64 load per fragment.

__device__ __forceinline__ v8f gemm_tile(const float* act, int ldA, int ksteps,
                                         const float* __restrict__ Wp, int Kpad,
                                         int n0, int lane) {
  const int m    = lane & 15;
  const int koff = (lane >> 4) << 1;
  const float* arow = act + m * ldA + koff;
  const float* brow = Wp + (n0 + m) * Kpad + koff;
  v8f c = {};
  for (int ks = 0; ks < ksteps; ++ks) {
    v2f a = *(const v2f*)(arow + ks * 4);
    v2f b = *(const v2f*)(brow + ks * 4);
    c = __builtin_amdgcn_wmma_f32_16x16x4_f32(false, a, false, b, (short)0, c,
                                              false, false);
  }
  return c;
}

__device__ __forceinline__ void store_act(float* out, int ldO, v8f c,
                                          const float* __restrict__ biasp,
                                          int n0, int lane, bool lrelu) {
  int col = lane & 15, rbase = (lane >> 4) << 3, n = n0 + col;
  float bv = biasp[n];                       // padded: 0 beyond valid cols
#pragma unroll
  for (int r = 0; r < 8; ++r) {
    float v = c[r] + bv;
    if (lrelu) v = (v >= 0.f) ? v : SLOPE * v;
    out[(r + rbase) * ldO + n] = v;
  }
}

// ---- Kernel A: L1..L4 for a 16-timestep tile ------------------------------

__global__ __launch_bounds__(32) void k_encode(
    const float* __restrict__ z_seq, const float* __restrict__ a_seq,
    const float* __restrict__ pad,
    float* __restrict__ u_out,      // [T_TOT][64]  (50 valid)
    float* __restrict__ inter_out)  // [T_TOT][112] (100 valid)
{
  __shared__ __align__(16) float X[16 * KP1];
  __shared__ __align__(16) float bufA[16 * NP];
  __shared__ __align__(16) float bufB[16 * NP];
  const int lane = threadIdx.x;
  const int t0   = blockIdx.x * 16;
  const float* W1p = pad + O_W1P; const float* W2p = pad + O_W2P;
  const float* W3p = pad + O_W3P; const float* W4p = pad + O_W4P;

  // Stage az = [a(2) | z(100)] -> 16x104. Row t=16383 is in-bounds (T=16384)
  // and its downstream u value is provably never consumed -> no row guard.
  for (int i = lane; i < 16 * KP1; i += 32) {
    int m = i / KP1, k = i % KP1, t = t0 + m;
    float v = 0.f;
    if (k < 2)        v = a_seq[t * 2 + k];
    else if (k < 102) v = z_seq[t * 100 + (k - 2)];
    X[i] = v;
  }
  __syncthreads();

  for (int nt = 0; nt < 7; ++nt)   // L1
    store_act(bufA, NP, gemm_tile(X, KP1, 26, W1p, KP1, nt * 16, lane),
              pad + O_B1P, nt * 16, lane, true);
  __syncthreads();
  for (int nt = 0; nt < 7; ++nt)   // L2 -> inter2
    store_act(bufB, NP, gemm_tile(bufA, NP, 28, W2p, NP, nt * 16, lane),
              pad + O_B2P, nt * 16, lane, true);
  __syncthreads();
  for (int nt = 0; nt < 7; ++nt)   // L3
    store_act(bufA, NP, gemm_tile(bufB, NP, 28, W3p, NP, nt * 16, lane),
              pad + O_B3P, nt * 16, lane, true);
  __syncthreads();

  // L4: u = bufA @ W4p^T + b4p (no activation), unconditional store
  const float* b4p = pad + O_B4P;
  for (int nt = 0; nt < 4; ++nt) {
    v8f c = gemm_tile(bufA, NP, 28, W4p, NP, nt * 16, lane);
    int col = lane & 15, rbase = (lane >> 4) << 3, n = nt * 16 + col;
    float bv = b4p[n];
#pragma unroll
    for (int r = 0; r < 8; ++r)
      u_out[(t0 + r + rbase) * 64 + n] = c[r] + bv;
  }

  for (int i = lane; i < 16 * NP; i += 32) {
    int m = i / NP;
    inter_out[(t0 + m) * NP + (i % NP)] = bufB[i];
  }
}

// ---- Scan: inclusive prefix sum of u -> h (128 chunks x 128 steps) --------

__global__ __launch_bounds__(64) void k_scan_partial(const float* __restrict__ u,
                                                     float* __restrict__ csum) {
  int c = threadIdx.x, blk = blockIdx.x;
  float s = 0.f;
  for (int i = 0; i < 128; ++i) s += u[(blk * 128 + i) * 64 + c];
  csum[blk * 64 + c] = s;
}

__global__ __launch_bounds__(64) void k_scan_offsets(const float* __restrict__ csum,
                                                     float* __restrict__ coff) {
  int c = threadIdx.x;
  float run = 0.f;
  for (int b = 0; b < 128; ++b) { coff[b * 64 + c] = run; run += csum[b * 64 + c]; }
}

__global__ __launch_bounds__(64) void k_scan_apply(const float* __restrict__ u,
                                                   const float* __restrict__ coff,
                                                   float* __restrict__ h) {
  int c = threadIdx.x, blk = blockIdx.x;
  float run = coff[blk * 64 + c];
  for (int i = 0; i < 128; ++i) {
    int t = blk * 128 + i;
    run += u[t * 64 + c];
    h[t * 64 + c] = run;
  }
}

// ---- Kernel C: head (L5, Lz, Lt) + fused deterministic loss partials ------

__global__ __launch_bounds__(32) void k_head(
    const float* __restrict__ z_seq, const float* __restrict__ term_seq,
    const float* __restrict__ h, const float* __restrict__ inter,
    const float* __restrict__ pad, const float* __restrict__ bt,
    float* __restrict__ part)  // [gridDim.x][2] : (sse, bce)
{
  __shared__ __align__(16) float inp[16 * KP5];
  __shared__ __align__(16) float i5[16 * NP];
  __shared__ float red[32];
  const int lane = threadIdx.x;
  const int t0   = blockIdx.x * 16;
  const float* W5p = pad + O_W5P; const float* Wzp = pad + O_WZP;
  const float* Wtp = pad + O_WTP;

  for (int i = lane; i < 16 * KP5; i += 32) {
    int m = i / KP5, k = i % KP5, t = t0 + m;
    float v = 0.f;
    if (k < 50)                  v = h[t * 64 + k];
    else if (k >= 64 && k < 164) v = inter[t * NP + (k - 64)];
    inp[i] = v;
  }
  __syncthreads();

  for (int nt = 0; nt < 7; ++nt)   // L5 (gap mapping baked into W5p)
    store_act(i5, NP, gemm_tile(inp, KP5, 44, W5p, KP5, nt * 16, lane),
              pad + O_B5P, nt * 16, lane, true);
  __syncthreads();

  // Lz + squared-error: clamped-index unconditional loads, select-only guards
  const float* bzp = pad + O_BZP;
  float sse = 0.f;
  for (int nt = 0; nt < 7; ++nt) {
    v8f c = gemm_tile(i5, NP, 28, Wzp, NP, nt * 16, lane);
    int col = lane & 15, rbase = (lane >> 4) << 3, n = nt * 16 + col;
    float bv = bzp[n];
    bool nv  = (n < 100);
    int  nn  = nv ? n : 0;
#pragma unroll
    for (int r = 0; r < 8; ++r) {
      int t  = t0 + r + rbase;
      int tt = (t < S_STEPS) ? t : (S_STEPS - 1);
      float d = z_seq[(tt + 1) * 100 + nn] - (c[r] + bv);
      sse += (nv && t < S_STEPS) ? d * d : 0.f;
    }
  }

  // Lt: scalar logit per row; all 32 lanes compute (16-31 duplicate), select
  float bce;
  {
    int m = lane & 15, t = t0 + m;
    float x = bt[0];
    for (int k = 0; k < 100; ++k) x += i5[m * NP + k] * Wtp[k];
    int tt = (t < S_STEPS) ? t : (S_STEPS - 1);
    float term = term_seq[tt + 1];
    float v = fmaxf(x, 0.f) - x * term + log1pf(expf(-fabsf(x)));
    bce = (lane < 16 && t < S_STEPS) ? v : 0.f;
  }

  // Deterministic per-tile reduction (bit-stable across graph replays)
  red[lane] = sse;  __syncthreads();
  if (lane == 0) { float s = 0.f; for (int i = 0; i < 32; ++i) s += red[i];
                   part[blockIdx.x * 2 + 0] = s; }
  __syncthreads();
  red[lane] = bce;  __syncthreads();
  if (lane == 0) { float s = 0.f; for (int i = 0; i < 32; ++i) s += red[i];
                   part[blockIdx.x * 2 + 1] = s; }
}

// ---- Finalize -------------------------------------------------------------

__global__ void k_finalize(const float* __restrict__ part, float* __restrict__ out) {
  if (threadIdx.x == 0 && blockIdx.x == 0) {
    float sse = 0.f, bce = 0.f;
    for (int i = 0; i < 1024; ++i) { sse += part[i * 2 + 0]; bce += part[i * 2 + 1]; }
    float zl = sse / (float)(S_STEPS * 100);
    float tl = bce / (float)S_STEPS;
    out[0] = zl + tl; out[1] = zl; out[2] = tl;
  }
}

// ---------------------------------------------------------------------------

extern "C" void kernel_launch(void* const* d_in, const int* in_sizes, int n_in,
                              void* d_out, int out_size, void* d_ws, size_t ws_size,
                              hipStream_t stream) {
  const float* z_seq    = (const float*)d_in[0];
  const float* a_seq    = (const float*)d_in[1];
  const float* term_seq = (const float*)d_in[2];
  const float* W1 = (const float*)d_in[3];  const float* b1 = (const float*)d_in[4];
  const float* W2 = (const float*)d_in[5];  const float* b2 = (const float*)d_in[6];
  const float* W3 = (const float*)d_in[7];  const float* b3 = (const float*)d_in[8];
  const float* W4 = (const float*)d_in[9];  const float* b4 = (const float*)d_in[10];
  const float* W5 = (const float*)d_in[11]; const float* b5 = (const float*)d_in[12];
  const float* Wz = (const float*)d_in[13]; const float* bz = (const float*)d_in[14];
  const float* Wt = (const float*)d_in[15]; const float* bt = (const float*)d_in[16];

  float* ws    = (float*)d_ws;
  float* u     = ws;                             // 16384*64
  float* h     = u     + (size_t)T_TOT * 64;     // 16384*64
  float* inter = h     + (size_t)T_TOT * 64;     // 16384*112
  float* csum  = inter + (size_t)T_TOT * NP;     // 128*64
  float* coff  = csum  + 128 * 64;               // 128*64
  float* part  = coff  + 128 * 64;               // 1024*2
  float* pad   = part  + 2048;                   // PAD_TOTAL

  k_pad_all<<<(PAD_TOTAL + 255) / 256, 256, 0, stream>>>(
      W1, W2, W3, W4, W5, Wz, Wt, b1, b2, b3, b4, b5, bz, pad);
  k_encode<<<1024, 32, 0, stream>>>(z_seq, a_seq, pad, u, inter);
  k_scan_partial<<<128, 64, 0, stream>>>(u, csum);
  k_scan_offsets<<<1,   64, 0, stream>>>(csum, coff);
  k_scan_apply  <<<128, 64, 0, stream>>>(u, coff, h);
  k_head<<<1024, 32, 0, stream>>>(z_seq, term_seq, h, inter, pad, bt, part);
  k_finalize<<<1, 32, 0, stream>>>(part, (float*)d_out);
}